// BiLSTMLM_70506183131220
// MI455X (gfx1250) — compile-verified
//
#include <hip/hip_runtime.h>
#include <hip/hip_bf16.h>
#include <stdint.h>

// ---------------------------------------------------------------------------
// BiLSTM LM for MI455X (gfx1250, wave32).
//  - embed gather kernel
//  - 2-block sequential LSTM scan; gate GEMMs via v_wmma_f32_16x16x4_f32
//    (K=48 -> 12 k-steps); weights live in registers for the whole scan;
//    x_{t+1} is double-buffered into LDS with global_load_async_to_lds_b128
//    (ASYNCcnt) overlapping step-t compute; branchless activations through a
//    single native v_tanh_f32 (sigmoid = affine-scaled tanh).
//  - projection GEMM (4096x32 @ 32x10000) via fp32 WMMA, K=32 -> 8 k-steps.
//  - fused log-softmax (register-resident rows) + zero tail rows.
// ---------------------------------------------------------------------------

typedef __attribute__((ext_vector_type(2))) float v2f;
typedef __attribute__((ext_vector_type(8))) float v8f;

#define S_LEN 256
#define BATCH 32
#define VOCAB 10000
#define EDIM  32
#define HDIM  16
#define NSTEP 255          // N = S-1
#define MHALF 128          // M = (N+1)/2
#define NTILES 625         // VOCAB / 16

__device__ __forceinline__ float fast_tanh(float x) {
#if __has_builtin(__builtin_amdgcn_tanhf)
    return __builtin_amdgcn_tanhf(x);           // native v_tanh_f32
#else
    float e = __expf(2.0f * x);
    return 1.0f - 2.0f * __builtin_amdgcn_rcpf(e + 1.0f);
#endif
}

// Async DMA: 16 B per lane, 512 B per wave-issue, tracked by ASYNCcnt.
__device__ __forceinline__ void async_copy_b128(const void* gsrc, void* lds_dst) {
    uint32_t ldsoff = (uint32_t)(uintptr_t)lds_dst;   // low 32b of generic = LDS addr
    uint64_t gaddr  = (uint64_t)(uintptr_t)gsrc;
    asm volatile("global_load_async_to_lds_b128 %0, %1, off"
                 :: "v"(ldsoff), "v"(gaddr) : "memory");
}

// ---------------------------------------------------------------------------
__global__ __launch_bounds__(256)
void embed_kernel(const int* __restrict__ idx, const float* __restrict__ emb_w,
                  float* __restrict__ emb_out) {
    int i = blockIdx.x * 256 + threadIdx.x;          // over S*B*E
    if (i < S_LEN * BATCH * EDIM) {
        int e  = i & (EDIM - 1);
        int tb = i >> 5;                              // / EDIM
        emb_out[i] = emb_w[idx[tb] * EDIM + e];
    }
}

// ---------------------------------------------------------------------------
// One block per direction (blockIdx.x: 0 = L->R, 1 = R->L). 8 waves:
// wave = 4*mtile + gate.  Gate tile (16 batch rows x 16 hidden) =
// comb(16x48) @ Wg^T(48x16) as 12 chained 16x16x4 fp32 WMMAs; k-steps 0..7
// read the async-filled x buffer, k-steps 8..11 read the h buffer.
__global__ __launch_bounds__(256)
void lstm_kernel(const float* __restrict__ emb,
                 const float* __restrict__ Wf, const float* __restrict__ bf,
                 const float* __restrict__ Wi, const float* __restrict__ bi,
                 const float* __restrict__ Wc, const float* __restrict__ bc,
                 const float* __restrict__ Wo, const float* __restrict__ bo,
                 const float* __restrict__ h0_lr, const float* __restrict__ c0_lr,
                 const float* __restrict__ h0_rl, const float* __restrict__ c0_rl,
                 float* __restrict__ concat)          // (MHALF, B, 2H)
{
    __shared__ float xbuf[2][BATCH][EDIM];            // double-buffered x_t (4 KB each)
    __shared__ float hbuf[BATCH][HDIM];
    __shared__ float cstate[BATCH][HDIM];
    __shared__ float gates[4][BATCH][HDIM];

    const int dir  = blockIdx.x;
    const int tid  = threadIdx.x;
    const int lane = tid & 31;
    const int wave = tid >> 5;
    const int gate = wave & 3;                        // 0=f 1=i 2=C~ 3=o
    const int mt   = wave >> 2;                       // batch-row tile (0/1)

    const float* Wg = (gate == 0) ? Wf : (gate == 1) ? Wi : (gate == 2) ? Wc : Wo;
    const float* bg = (gate == 0) ? bf : (gate == 1) ? bi : (gate == 2) ? bc : bo;

    // Branchless activation: act(g) = so * tanh(si * g) + off
    //   C~ gate: tanh      -> (1, 1, 0)
    //   f/i/o  : sigmoid   -> (0.5, 0.5, 0.5)   [sigmoid(x)=0.5*tanh(x/2)+0.5]
    const float act_si  = (gate == 2) ? 1.0f : 0.5f;
    const float act_so  = (gate == 2) ? 1.0f : 0.5f;
    const float act_off = (gate == 2) ? 0.0f : 0.5f;

    // B-operand fragments held for the whole scan: B[k][n] = Wg[n][k].
    // Lane L: column n = L%16; k = 4*s + 2*(L/16) + component.
    const int ncol = lane & 15;
    const int koff = (lane >> 4) * 2;
    v2f wfrag[12];
#pragma unroll
    for (int s = 0; s < 12; ++s) {
        wfrag[s].x = Wg[ncol * 48 + s * 4 + koff];
        wfrag[s].y = Wg[ncol * 48 + s * 4 + koff + 1];
    }
    const float biasn = bg[ncol];

    // initial state
    const float* h0 = dir ? h0_rl : h0_lr;
    const float* c0 = dir ? c0_rl : c0_lr;
    for (int i = tid; i < BATCH * HDIM; i += 256) {
        int b = i >> 4, h = i & 15;
        hbuf[b][h]   = h0[i];
        cstate[b][h] = c0[i];
    }

    // LR needs hidden-before-step for steps 0..127 -> run 128 steps.
    // RL needs hidden-before-step for steps 127..254 -> run 255 steps.
    const int T = dir ? NSTEP : MHALF;

    // Per-wave async chunk: wave w moves bytes [512w, 512w+512) of x_t.
    const int chunk = wave * 512 + lane * 16;
    {   // prefetch x_0
        const int xt0 = dir ? (S_LEN - 1) : 0;
        async_copy_b128((const char*)(emb + (size_t)xt0 * BATCH * EDIM) + chunk,
                        (char*)&xbuf[0][0][0] + chunk);
    }

    for (int t = 0; t < T; ++t) {
        const int buf = t & 1;

        // ---- phase 1: prefetch x_{t+1}; emit hidden-before-step; wait x_t ----
        if (t + 1 < T) {
            const int xt = dir ? (S_LEN - 2 - t) : (t + 1);
            async_copy_b128((const char*)(emb + (size_t)xt * BATCH * EDIM) + chunk,
                            (char*)&xbuf[buf ^ 1][0][0] + chunk);
        }

        const bool doStore = dir ? (t >= MHALF - 1) : (t < MHALF);
        if (doStore) {
            const int j = dir ? (NSTEP - 1 - t) : t;  // hRL[j] = hRL_old[254-t]
            float* dst = concat + (size_t)j * BATCH * 2 * HDIM + (dir ? HDIM : 0);
            for (int i = tid; i < BATCH * HDIM; i += 256) {
                int b = i >> 4, h = i & 15;
                dst[b * 2 * HDIM + h] = hbuf[b][h];
            }
        }

        if (t + 1 < T) {                 // 2 outstanding: t (oldest) + t+1
            asm volatile("s_wait_asynccnt 1" ::: "memory");
        } else {                         // only t outstanding
            asm volatile("s_wait_asynccnt 0" ::: "memory");
        }
        __syncthreads();                 // publish x_t chunks + gate h reuse

        // ---- phase 2: gate GEMM (WMMA) + activation ----
        v8f acc;
#pragma unroll
        for (int v = 0; v < 8; ++v) acc[v] = biasn;   // bias broadcast per column
        {
            const int arow = mt * 16 + ncol;          // A: lane L -> row m = L%16
#pragma unroll
            for (int s = 0; s < 8; ++s) {             // k = 0..31 from x buffer
                v2f a;
                a.x = xbuf[buf][arow][s * 4 + koff];
                a.y = xbuf[buf][arow][s * 4 + koff + 1];
                acc = __builtin_amdgcn_wmma_f32_16x16x4_f32(
                    false, a, false, wfrag[s], (short)0, acc, false, false);
            }
#pragma unroll
            for (int s = 8; s < 12; ++s) {            // k = 32..47 from h buffer
                v2f a;
                a.x = hbuf[arow][(s - 8) * 4 + koff];
                a.y = hbuf[arow][(s - 8) * 4 + koff + 1];
                acc = __builtin_amdgcn_wmma_f32_16x16x4_f32(
                    false, a, false, wfrag[s], (short)0, acc, false, false);
            }
        }
        {
            const int rbase = mt * 16 + 8 * (lane >> 4);  // C/D: row = v + 8*(L/16)
#pragma unroll
            for (int v = 0; v < 8; ++v) {
                float g = fmaf(fast_tanh(acc[v] * act_si), act_so, act_off);
                gates[gate][rbase + v][ncol] = g;
            }
        }
        __syncthreads();

        // ---- phase 3: state update ----
        for (int i = tid; i < BATCH * HDIM; i += 256) {
            int b = i >> 4, h = i & 15;
            float f  = gates[0][b][h];
            float ig = gates[1][b][h];
            float ct = gates[2][b][h];
            float o  = gates[3][b][h];
            float cn = f * cstate[b][h] + ig * ct;
            cstate[b][h] = cn;
            hbuf[b][h]   = o * fast_tanh(cn);
        }
        __syncthreads();
    }
}

// ---------------------------------------------------------------------------
// logits = concat(4096x32) @ h2o_w^T(32x10000) + b, fp32 WMMA, K=32 = 8 steps.
// Block: 16 rows x 128 cols (8 waves, one 16x16 N-tile each).
__global__ __launch_bounds__(256)
void logits_kernel(const float* __restrict__ concat,   // (MHALF*B, 2H)
                   const float* __restrict__ h2o_w,    // (V, 2H)
                   const float* __restrict__ h2o_b,    // (V)
                   float* __restrict__ out)            // (S*B, V)
{
    const int tid  = threadIdx.x;
    const int lane = tid & 31;
    const int wave = tid >> 5;
    const int nt   = blockIdx.y * 8 + wave;            // N tile
    if (nt >= NTILES) return;                          // wave-uniform exit

    const int mrow0 = blockIdx.x * 16;
    const int ncol  = lane & 15;
    const int koff  = (lane >> 4) * 2;
    const int n     = nt * 16 + ncol;

    v8f acc;
    const float bn = h2o_b[n];
#pragma unroll
    for (int v = 0; v < 8; ++v) acc[v] = bn;

    const float* arow = concat + (size_t)(mrow0 + ncol) * (2 * HDIM);
    const float* brow = h2o_w + (size_t)n * (2 * HDIM);
#pragma unroll
    for (int s = 0; s < 8; ++s) {
        v2f a, b;
        a.x = arow[s * 4 + koff]; a.y = arow[s * 4 + koff + 1];
        b.x = brow[s * 4 + koff]; b.y = brow[s * 4 + koff + 1];
        acc = __builtin_amdgcn_wmma_f32_16x16x4_f32(
            false, a, false, b, (short)0, acc, false, false);
    }

    const int rbase = mrow0 + 8 * (lane >> 4);
#pragma unroll
    for (int v = 0; v < 8; ++v)
        out[(size_t)(rbase + v) * VOCAB + n] = acc[v];
}

// ---------------------------------------------------------------------------
// One block per output row. Rows >= MHALF*B are zero-filled (reference pads).
// Whole 10000-wide row kept in 40 regs/lane: read once, write once.
__global__ __launch_bounds__(256)
void logsoftmax_kernel(float* __restrict__ out) {
    const int row = blockIdx.x;                        // 0 .. S*B-1
    const int tid = threadIdx.x;
    float* p = out + (size_t)row * VOCAB;

    if (row >= MHALF * BATCH) {
        for (int i = tid; i < VOCAB; i += 256) p[i] = 0.0f;
        return;
    }

    __shared__ float red[256];
    float v[40];
    float mx = -INFINITY;
#pragma unroll
    for (int c = 0; c < 40; ++c) {
        int i = tid + c * 256;
        v[c] = (i < VOCAB) ? p[i] : -INFINITY;
        mx = fmaxf(mx, v[c]);
    }
    red[tid] = mx;
    __syncthreads();
    for (int s = 128; s > 0; s >>= 1) {
        if (tid < s) red[tid] = fmaxf(red[tid], red[tid + s]);
        __syncthreads();
    }
    mx = red[0];
    __syncthreads();

    float sum = 0.0f;
#pragma unroll
    for (int c = 0; c < 40; ++c) {
        int i = tid + c * 256;
        sum += (i < VOCAB) ? __expf(v[c] - mx) : 0.0f;
    }
    red[tid] = sum;
    __syncthreads();
    for (int s = 128; s > 0; s >>= 1) {
        if (tid < s) red[tid] += red[tid + s];
        __syncthreads();
    }
    const float lse = mx + __logf(red[0]);

#pragma unroll
    for (int c = 0; c < 40; ++c) {
        int i = tid + c * 256;
        if (i < VOCAB) p[i] = v[c] - lse;
    }
}

// ---------------------------------------------------------------------------
extern "C" void kernel_launch(void* const* d_in, const int* in_sizes, int n_in,
                              void* d_out, int out_size, void* d_ws, size_t ws_size,
                              hipStream_t stream) {
    (void)in_sizes; (void)n_in; (void)out_size; (void)ws_size;

    const int*   idx   = (const int*)  d_in[0];
    const float* embw  = (const float*)d_in[1];
    const float* Wf    = (const float*)d_in[2];  const float* bf = (const float*)d_in[3];
    const float* Wi    = (const float*)d_in[4];  const float* bi = (const float*)d_in[5];
    const float* Wc    = (const float*)d_in[6];  const float* bc = (const float*)d_in[7];
    const float* Wo    = (const float*)d_in[8];  const float* bo = (const float*)d_in[9];
    const float* h2o_w = (const float*)d_in[10]; const float* h2o_b = (const float*)d_in[11];
    const float* h0_lr = (const float*)d_in[12]; const float* c0_lr = (const float*)d_in[13];
    const float* h0_rl = (const float*)d_in[14]; const float* c0_rl = (const float*)d_in[15];
    float* out = (float*)d_out;

    // workspace layout (fp32): emb (S*B*E) | concat (MHALF*B*2H)  ~1.6 MB
    float* emb    = (float*)d_ws;
    float* concat = emb + S_LEN * BATCH * EDIM;

    embed_kernel<<<(S_LEN * BATCH * EDIM + 255) / 256, 256, 0, stream>>>(idx, embw, emb);

    lstm_kernel<<<2, 256, 0, stream>>>(emb, Wf, bf, Wi, bi, Wc, bc, Wo, bo,
                                       h0_lr, c0_lr, h0_rl, c0_rl, concat);

    logits_kernel<<<dim3(MHALF * BATCH / 16, (NTILES + 7) / 8), 256, 0, stream>>>(
        concat, h2o_w, h2o_b, out);

    logsoftmax_kernel<<<S_LEN * BATCH, 256, 0, stream>>>(out);
}